// BatchedRegionProposalNetwork_45698452030231
// MI455X (gfx1250) — compile-verified
//
#include <hip/hip_runtime.h>
#include <stdint.h>

#define PRE_N   1000
#define POST_N  1000
#define SORT_N  1024
#define NMS_W   32          // 32 x u32 = 1024 bits per row
#define CH_ROWS 128         // rows per double-buffered mask chunk (16KB per buffer)
#define NMS_THRESH 0.7f
#define MIN_SIZE   0.001f

#if defined(__HIP_DEVICE_COMPILE__) && __has_builtin(__builtin_amdgcn_global_load_async_to_lds_b128)
#define ASYNC_LDS_OK 1
#endif
#if defined(__HIP_DEVICE_COMPILE__) && __has_builtin(__builtin_amdgcn_s_wait_asynccnt)
#define ASYNC_WAIT_OK 1
#endif

typedef __attribute__((ext_vector_type(4))) int int4v;
#define AS1(p) ((__attribute__((address_space(1))) int4v*)(p))
#define AS3(p) ((__attribute__((address_space(3))) int4v*)(p))

// ---------- order-preserving float <-> uint key ----------
__device__ __forceinline__ unsigned keyOf(float f) {
    unsigned u = __float_as_uint(f);
    return u ^ (unsigned)((((int)u) >> 31) | 0x80000000);
}
__device__ __forceinline__ float keyInv(unsigned k) {
    unsigned u = (k & 0x80000000u) ? (k ^ 0x80000000u) : ~k;
    return __uint_as_float(u);
}

// ---------- K0: zero ----------
__global__ void k_zero(uint32_t* p, long long n) {
    long long i = (long long)blockIdx.x * blockDim.x + threadIdx.x;
    if (i < n) p[i] = 0u;
}

// ---------- K1/K3: 16-bit histogram passes ----------
__global__ void k_hist(const float* __restrict__ obj, int N, uint32_t* __restrict__ hist,
                       int pass, const uint32_t* __restrict__ b1arr) {
    int b = blockIdx.y;
    const float* src = obj + (long long)b * N;
    uint32_t* h = hist + (long long)b * 65536;
    int stride = gridDim.x * blockDim.x;
    uint32_t want = (pass == 1) ? b1arr[b] : 0u;
    for (int i = blockIdx.x * blockDim.x + threadIdx.x; i < N; i += stride) {
        __builtin_prefetch(src + i + 2 * stride, 0, 0);   // global_prefetch_b8
        unsigned k = keyOf(src[i]);
        if (pass == 0) {
            atomicAdd(&h[k >> 16], 1u);
        } else if ((k >> 16) == want) {
            atomicAdd(&h[k & 0xFFFFu], 1u);
        }
    }
}

// ---------- K2/K4: find threshold bin from 65536-bin histogram ----------
__global__ void k_select(const uint32_t* __restrict__ hist, int pass,
                         uint32_t* b1, uint32_t* g1,
                         uint32_t* selT, uint32_t* selG, uint32_t* selNeed) {
    int b = blockIdx.x;
    const uint32_t* h = hist + (long long)b * 65536;
    int t = threadIdx.x;                    // 1024 threads, 64 bins each
    __shared__ uint32_t s[1024];
    uint32_t K = (pass == 0) ? (uint32_t)PRE_N : (uint32_t)PRE_N - g1[b];
    int base = t * 64;
    uint32_t sum = 0;
    for (int i = 0; i < 64; ++i) sum += h[base + i];
    s[t] = sum;
    __syncthreads();
    uint32_t above = 0;                     // count in bins strictly above this range
    for (int u = t + 1; u < 1024; ++u) above += s[u];
    if (above < K && above + sum >= K) {    // exactly one thread
        uint32_t c = above;
        for (int bin = base + 63; bin >= base; --bin) {
            uint32_t hv = h[bin];
            if (c + hv >= K) {
                if (pass == 0) { b1[b] = (uint32_t)bin; g1[b] = c; }
                else {
                    uint32_t T = (b1[b] << 16) | (uint32_t)bin;
                    uint32_t G = g1[b] + c;
                    selT[b] = T; selG[b] = G; selNeed[b] = (uint32_t)PRE_N - G;
                }
                break;
            }
            c += hv;
        }
    }
}

// ---------- K5: per-block tie counts (index order) ----------
__global__ void k_tiecount(const float* __restrict__ obj, int N,
                           const uint32_t* __restrict__ selT,
                           uint32_t* __restrict__ blockTie, int NB) {
    int b = blockIdx.y;
    int i = blockIdx.x * blockDim.x + threadIdx.x;
    bool tie = false;
    if (i < N) tie = (keyOf(obj[(long long)b * N + i]) == selT[b]);
    int cnt = __syncthreads_count(tie);
    if (threadIdx.x == 0) blockTie[(long long)b * NB + blockIdx.x] = (uint32_t)cnt;
}

// ---------- K6: exclusive scan of tie counts ----------
__global__ void k_scantie(const uint32_t* __restrict__ blockTie,
                          uint32_t* __restrict__ blockOff, int NB) {
    int b = blockIdx.x;
    if (threadIdx.x == 0) {
        uint32_t acc = 0;
        for (int i = 0; i < NB; ++i) {
            blockOff[(long long)b * NB + i] = acc;
            acc += blockTie[(long long)b * NB + i];
        }
    }
}

// ---------- K7: emit selected (key, idx) pairs ----------
__global__ void k_emit(const float* __restrict__ obj, int N,
                       const uint32_t* __restrict__ selT, const uint32_t* __restrict__ selG,
                       const uint32_t* __restrict__ selNeed,
                       const uint32_t* __restrict__ blockOff, int NB,
                       uint32_t* __restrict__ cnt, unsigned long long* __restrict__ sel) {
    int b = blockIdx.y;
    int i = blockIdx.x * blockDim.x + threadIdx.x;
    uint32_t T = selT[b], G = selG[b], Need = selNeed[b];
    unsigned k = 0;
    bool inb = (i < N);
    if (inb) k = keyOf(obj[(long long)b * N + i]);
    bool gt  = inb && (k > T);
    bool tie = inb && (k == T);
    unsigned long long V = ((unsigned long long)k << 32) | (uint32_t)(~(uint32_t)i);
    if (gt) {                               // order among strictly-greater fixed later by sort
        uint32_t slot = atomicAdd(&cnt[b], 1u);
        sel[(long long)b * SORT_N + slot] = V;
    }
    // ties: include only the Need smallest indices (index-ordered rank)
    unsigned lane = threadIdx.x & 31u;
    unsigned wave = threadIdx.x >> 5;
    unsigned m = (unsigned)__ballot(tie);   // wave32: low 32 bits valid
    __shared__ uint32_t wtot[8];
    if (lane == 0) wtot[wave] = (uint32_t)__popc(m);
    __syncthreads();
    if (tie) {
        uint32_t off = blockOff[(long long)b * NB + blockIdx.x];
        for (unsigned w2 = 0; w2 < wave; ++w2) off += wtot[w2];
        off += (uint32_t)__popc(m & ((1u << lane) - 1u));
        if (off < Need) sel[(long long)b * SORT_N + G + off] = V;
    }
}

// ---------- K8: bitonic sort 1024 packed values, descending ----------
__global__ void k_sort(unsigned long long* __restrict__ sel) {
    int b = blockIdx.x;
    int t = threadIdx.x;                    // 1024 threads
    __shared__ unsigned long long a[SORT_N];
    a[t] = sel[(long long)b * SORT_N + t];
    __syncthreads();
    for (unsigned k = 2; k <= SORT_N; k <<= 1) {
        for (unsigned j = k >> 1; j > 0; j >>= 1) {
            unsigned ixj = (unsigned)t ^ j;
            if (ixj > (unsigned)t) {
                unsigned long long x = a[t], y = a[ixj];
                bool descSeg = ((t & k) == 0);
                if (descSeg ? (x < y) : (x > y)) { a[t] = y; a[ixj] = x; }
            }
            __syncthreads();
        }
    }
    sel[(long long)b * SORT_N + t] = a[t];
}

// ---------- K9: gather(async to LDS) + clip + level offset + suppression bit-matrix ----------
__global__ void k_mask(const float* __restrict__ proposals, const int* __restrict__ levels, int N,
                       const int* __restrict__ imgh, const int* __restrict__ imgw,
                       const unsigned long long* __restrict__ sel, uint32_t* __restrict__ maskOut) {
    __shared__ __align__(16) float raw[PRE_N * 4];
    __shared__ float ox1[PRE_N], oy1[PRE_N], ox2[PRE_N], oy2[PRE_N], areaS[PRE_N];
    __shared__ float red[1024];
    __shared__ float mval;
    int b = blockIdx.x;
    int t = threadIdx.x;                    // 1024 threads
    float W = (float)imgw[0], H = (float)imgh[0];
    unsigned idx = 0;
    if (t < PRE_N) {
        unsigned long long V = sel[(long long)b * SORT_N + t];
        idx = ~((uint32_t)V);
        const float* gsrc = proposals + ((long long)b * N + idx) * 4;  // 16B aligned
#if defined(ASYNC_LDS_OK)
        __builtin_amdgcn_global_load_async_to_lds_b128(AS1(gsrc), AS3(&raw[t * 4]), 0, 0);
#else
        float4 v = *(const float4*)gsrc;
        *(float4*)&raw[t * 4] = v;
#endif
    }
#if defined(ASYNC_WAIT_OK) && defined(ASYNC_LDS_OK)
    __builtin_amdgcn_s_wait_asynccnt(0);
#endif
    __syncthreads();
    float lmax = -1e30f;
    if (t < PRE_N) {
        float x1 = fminf(fmaxf(raw[t * 4 + 0], 0.f), W);
        float y1 = fminf(fmaxf(raw[t * 4 + 1], 0.f), H);
        float x2 = fminf(fmaxf(raw[t * 4 + 2], 0.f), W);
        float y2 = fminf(fmaxf(raw[t * 4 + 3], 0.f), H);
        ox1[t] = x1; oy1[t] = y1; ox2[t] = x2; oy2[t] = y2;
        areaS[t] = fmaxf(x2 - x1, 0.f) * fmaxf(y2 - y1, 0.f);
        lmax = fmaxf(fmaxf(x1, y1), fmaxf(x2, y2));
    }
    red[t] = lmax;
    __syncthreads();
    for (int s2 = 512; s2 > 0; s2 >>= 1) {
        if (t < s2) red[t] = fmaxf(red[t], red[t + s2]);
        __syncthreads();
    }
    if (t == 0) mval = red[0] + 1.0f;
    __syncthreads();
    if (t < PRE_N) {
        float off = (float)levels[idx] * mval;      // batched-NMS level offset
        ox1[t] += off; oy1[t] += off; ox2[t] += off; oy2[t] += off;
    }
    __syncthreads();
    uint32_t* mrow = maskOut + (long long)b * PRE_N * NMS_W;
    for (int p = t; p < PRE_N * NMS_W; p += 1024) {
        int i = p >> 5, w = p & 31;
        float xi1 = ox1[i], yi1 = oy1[i], xi2 = ox2[i], yi2 = oy2[i], ai = areaS[i];
        uint32_t bits = 0;
        int j0 = w * 32;
        #pragma unroll 4
        for (int bj = 0; bj < 32; ++bj) {
            int j = j0 + bj;
            if (j < PRE_N && j > i) {
                float lx = fmaxf(xi1, ox1[j]);
                float ly = fmaxf(yi1, oy1[j]);
                float rx = fminf(xi2, ox2[j]);
                float ry = fminf(yi2, oy2[j]);
                float iw = fmaxf(rx - lx, 0.f);
                float ih = fmaxf(ry - ly, 0.f);
                float inter = iw * ih;
                float iou = inter / (ai + areaS[j] - inter + 1e-9f);
                if (iou > NMS_THRESH) bits |= (1u << bj);
            }
        }
        mrow[p] = bits;
    }
}

// ---------- K10: serial greedy suppression, double-buffered async mask staging + pack ----------
__global__ void k_final(const float* __restrict__ proposals, int N,
                        const int* __restrict__ imgh, const int* __restrict__ imgw,
                        const unsigned long long* __restrict__ sel,
                        const uint32_t* __restrict__ maskIn,
                        float* __restrict__ outBoxes, float* __restrict__ outScores) {
    __shared__ float bx1[PRE_N], by1[PRE_N], bx2[PRE_N], by2[PRE_N], sc[PRE_N];
    __shared__ uint32_t keep[NMS_W], pref[NMS_W];
    __shared__ __align__(16) uint32_t mbuf[2][CH_ROWS * NMS_W];   // 2 x 16KB
    int b = blockIdx.x;
    int t = threadIdx.x;                            // 1024 threads
    float W = (float)imgw[0], H = (float)imgh[0];
    const uint32_t* mrow = maskIn + (long long)b * PRE_N * NMS_W;
    const int nch = (PRE_N + CH_ROWS - 1) / CH_ROWS;

    // async (or fallback) copy of one mask chunk into a LDS buffer: 1 B128 per thread
    auto issueChunk = [&](int c, int bufsel) {
        int rows = PRE_N - c * CH_ROWS;
        if (rows > CH_ROWS) rows = CH_ROWS;
        int nld = (rows * NMS_W) >> 2;              // number of 16B transfers
        if (t < nld) {
            const uint32_t* g = mrow + (long long)c * CH_ROWS * NMS_W + t * 4;
#if defined(ASYNC_LDS_OK)
            __builtin_amdgcn_global_load_async_to_lds_b128(AS1(g), AS3(&mbuf[bufsel][t * 4]), 0, 0);
#else
            *(int4v*)&mbuf[bufsel][t * 4] = *(const int4v*)g;
#endif
        }
    };

    if (t < NMS_W) keep[t] = 0u;
    issueChunk(0, 0);                               // overlap chunk0 DMA with the gather below
    __syncthreads();                                // keep[] zeroed before atomicOr
    if (t < PRE_N) {
        unsigned long long V = sel[(long long)b * SORT_N + t];
        unsigned key = (unsigned)(V >> 32);
        unsigned idx = ~((uint32_t)V);
        const float* gp = proposals + ((long long)b * N + idx) * 4;
        float x1 = fminf(fmaxf(gp[0], 0.f), W);
        float y1 = fminf(fmaxf(gp[1], 0.f), H);
        float x2 = fminf(fmaxf(gp[2], 0.f), W);
        float y2 = fminf(fmaxf(gp[3], 0.f), H);
        bx1[t] = x1; by1[t] = y1; bx2[t] = x2; by2[t] = y2;
        sc[t] = keyInv(key);
        bool valid = ((x2 - x1) > MIN_SIZE) && ((y2 - y1) > MIN_SIZE);
        if (valid) atomicOr(&keep[t >> 5], 1u << (t & 31));
    }
#if defined(ASYNC_WAIT_OK) && defined(ASYNC_LDS_OK)
    __builtin_amdgcn_s_wait_asynccnt(0);
#endif
    __syncthreads();
    uint32_t kw = keep[t & 31];                     // wave0 lanes own keep word = lane
    for (int c = 0; c < nch; ++c) {
        if (c + 1 < nch) issueChunk(c + 1, (c + 1) & 1);   // DMA next chunk while wave0 works
        if (t < 32) {                               // single wave32 serial greedy loop (LDS)
            int r0 = c * CH_ROWS;
            int rows = PRE_N - r0;
            if (rows > CH_ROWS) rows = CH_ROWS;
            const uint32_t* mb = &mbuf[c & 1][0];
            for (int i = r0; i < r0 + rows; ++i) {
                uint32_t wv = __shfl(kw, i >> 5, 32);
                if ((wv >> (i & 31)) & 1u)
                    kw &= ~mb[(i - r0) * NMS_W + t];
            }
        }
#if defined(ASYNC_WAIT_OK) && defined(ASYNC_LDS_OK)
        __builtin_amdgcn_s_wait_asynccnt(0);
#endif
        __syncthreads();
    }
    if (t < 32) keep[t] = kw;
    __syncthreads();
    if (t == 0) {
        uint32_t acc = 0;
        for (int w = 0; w < NMS_W; ++w) { pref[w] = acc; acc += (uint32_t)__popc(keep[w]); }
    }
    __syncthreads();
    if (t < PRE_N) {
        uint32_t kwd = keep[t >> 5];
        if ((kwd >> (t & 31)) & 1u) {
            uint32_t r = pref[t >> 5] + (uint32_t)__popc(kwd & ((1u << (t & 31)) - 1u));
            float* ob = outBoxes + ((long long)b * POST_N + r) * 4;
            ob[0] = bx1[t]; ob[1] = by1[t]; ob[2] = bx2[t]; ob[3] = by2[t];
            outScores[(long long)b * POST_N + r] = sc[t];
        }
    }
}

extern "C" void kernel_launch(void* const* d_in, const int* in_sizes, int n_in,
                              void* d_out, int out_size, void* d_ws, size_t ws_size,
                              hipStream_t stream) {
    const float* proposals  = (const float*)d_in[0];   // [B,N,4]
    const float* objectness = (const float*)d_in[1];   // [B,N]
    const int*   levels     = (const int*)d_in[2];     // [N]
    const int*   imgh       = (const int*)d_in[3];
    const int*   imgw       = (const int*)d_in[4];
    float* out = (float*)d_out;

    const int N  = in_sizes[2];
    const int Bb = in_sizes[1] / N;
    const int NB = (N + 255) / 256;

    char* ws = (char*)d_ws;
    uint32_t* hist1 = (uint32_t*)ws;                                  // B*65536
    uint32_t* hist2 = hist1 + (size_t)Bb * 65536;                     // B*65536
    uint32_t* meta  = hist2 + (size_t)Bb * 65536;
    uint32_t* b1 = meta, *g1 = meta + Bb, *selT = meta + 2 * Bb,
            *selG = meta + 3 * Bb, *selNeed = meta + 4 * Bb, *cnt = meta + 5 * Bb;
    uint32_t* blockTie = meta + 6 * Bb;
    uint32_t* blockOff = blockTie + (size_t)Bb * NB;
    uintptr_t selAddr = (uintptr_t)(blockOff + (size_t)Bb * NB);
    selAddr = (selAddr + 7) & ~(uintptr_t)7;
    unsigned long long* sel = (unsigned long long*)selAddr;
    uint32_t* maskBuf = (uint32_t*)(sel + (size_t)Bb * SORT_N);

    float* outBoxes  = out;
    float* outScores = out + (size_t)Bb * POST_N * 4;

    long long zn = (long long)(((char*)(sel + (size_t)Bb * SORT_N)) - ws) / 4;
    k_zero<<<dim3((unsigned)((zn + 1023) / 1024)), 1024, 0, stream>>>(hist1, zn);
    k_zero<<<dim3((unsigned)((out_size + 1023) / 1024)), 1024, 0, stream>>>((uint32_t*)out, out_size);

    k_hist<<<dim3(128, Bb), 256, 0, stream>>>(objectness, N, hist1, 0, b1);
    k_select<<<Bb, 1024, 0, stream>>>(hist1, 0, b1, g1, selT, selG, selNeed);
    k_hist<<<dim3(128, Bb), 256, 0, stream>>>(objectness, N, hist2, 1, b1);
    k_select<<<Bb, 1024, 0, stream>>>(hist2, 1, b1, g1, selT, selG, selNeed);

    k_tiecount<<<dim3(NB, Bb), 256, 0, stream>>>(objectness, N, selT, blockTie, NB);
    k_scantie<<<Bb, 32, 0, stream>>>(blockTie, blockOff, NB);
    k_emit<<<dim3(NB, Bb), 256, 0, stream>>>(objectness, N, selT, selG, selNeed,
                                             blockOff, NB, cnt, sel);
    k_sort<<<Bb, 1024, 0, stream>>>(sel);
    k_mask<<<Bb, 1024, 0, stream>>>(proposals, levels, N, imgh, imgw, sel, maskBuf);
    k_final<<<Bb, 1024, 0, stream>>>(proposals, N, imgh, imgw, sel, maskBuf,
                                     outBoxes, outScores);
}